// AutoLink_l2_33998961116066
// MI455X (gfx1250) — compile-verified
//
#include <hip/hip_runtime.h>

#define D 128

typedef __attribute__((ext_vector_type(2))) float v2f;
typedef __attribute__((ext_vector_type(8))) float v8f;

// ---------------- degree / normalization ----------------

__global__ void deg_init_kernel(int* __restrict__ deg, int n) {
    int i = blockIdx.x * blockDim.x + threadIdx.x;
    if (i < n) deg[i] = 1;               // self-loop contributes 1
}

__global__ void deg_acc_kernel(const int* __restrict__ col, int* __restrict__ deg, int e) {
    int i = blockIdx.x * blockDim.x + threadIdx.x;
    if (i < e) atomicAdd(&deg[col[i]], 1);
}

__global__ void dis_kernel(const int* __restrict__ deg, float* __restrict__ dis, int n) {
    int i = blockIdx.x * blockDim.x + threadIdx.x;
    if (i < n) dis[i] = rsqrtf((float)deg[i]);   // deg >= 1 always
}

// ---------------- fp32 WMMA GEMM: hout[N,128] = act(hin) @ W ----------------
// Block = 256 threads = 8 waves. Block covers 16 rows x 128 cols;
// each wave owns one 16x16 tile, K=128 swept in steps of 4 via
// v_wmma_f32_16x16x4_f32 (32 WMMAs per wave, fully unrolled).

__global__ __launch_bounds__(256)
void gemm_wmma_kernel(const float* __restrict__ hin, long in_stride, int do_relu,
                      const float* __restrict__ W, float* __restrict__ hout, int nrows) {
    // 132-float row pitch: bank = (c + 132*r) % 64 -> rows 0..15 land on
    // distinct banks for a fixed column (132 % 64 == 4).
    __shared__ float As[16 * 132];

    const int row0 = blockIdx.x * 16;

    for (int i = threadIdx.x; i < 16 * D; i += 256) {
        int r = i >> 7, c = i & 127;
        int gr = row0 + r;
        float v = (gr < nrows) ? hin[(long)gr * in_stride + c] : 0.0f;
        if (do_relu) v = fmaxf(v, 0.0f);
        As[r * 132 + c] = v;
    }
    __syncthreads();

    const int lane    = threadIdx.x & 31;
    const int half    = lane >> 4;        // 0: K=0,1   1: K=2,3
    const int lr      = lane & 15;        // row (A) / col (B) within tile
    const int colbase = (threadIdx.x >> 5) * 16;

    v8f acc = {0.f, 0.f, 0.f, 0.f, 0.f, 0.f, 0.f, 0.f};

#pragma unroll
    for (int k = 0; k < D; k += 4) {
        v2f a, b;
        // A: 16x4 f32 fragment (lanes 0-15: K=k,k+1; lanes 16-31: K=k+2,k+3)
        a.x = As[lr * 132 + k + half * 2];
        a.y = As[lr * 132 + k + half * 2 + 1];
        // B: 4x16 f32 fragment, N striped across lanes, W row-major [K][N]
        b.x = W[(k + half * 2) * D + colbase + lr];
        b.y = W[(k + half * 2 + 1) * D + colbase + lr];
        acc = __builtin_amdgcn_wmma_f32_16x16x4_f32(
                  /*neg_a=*/false, a, /*neg_b=*/false, b,
                  /*c_mod=*/(short)0, acc, /*reuse_a=*/false, /*reuse_b=*/false);
    }

    // D layout: VGPR i -> row = half*8 + i, col = lr
#pragma unroll
    for (int i = 0; i < 8; i++) {
        int r = row0 + half * 8 + i;
        if (r < nrows) hout[(long)r * D + colbase + lr] = acc[i];
    }
}

// ---------------- self-loop + bias (initializes out layer slice) ----------------
// out[n, layer, d] = h'[n,d] * dis[n]^2 + b[d]

__global__ __launch_bounds__(256)
void selfloop_bias_kernel(const float* __restrict__ hp, const float* __restrict__ dis,
                          const float* __restrict__ bias, float* __restrict__ out,
                          int layer, int n) {
    int node = blockIdx.x * 2 + (threadIdx.x >> 7);
    int d = threadIdx.x & 127;
    if (node < n) {
        float s = dis[node];
        s = s * s;
        out[(long)node * 384 + layer * 128 + d] = hp[(long)node * 128 + d] * s + bias[d];
    }
}

// ---------------- edge scatter: one wave per edge ----------------
// out[row, layer, :] += h'[col, :] * dis[row]*dis[col]

__global__ __launch_bounds__(256)
void scatter_kernel(const int* __restrict__ rows, const int* __restrict__ cols,
                    const float* __restrict__ hp, const float* __restrict__ dis,
                    float* __restrict__ out, int layer, int e) {
    int eid = blockIdx.x * 8 + (threadIdx.x >> 5);
    if (eid >= e) return;
    int lane = threadIdx.x & 31;

    int r = rows[eid];
    int c = cols[eid];
    float nrm = dis[r] * dis[c];

    const float4* src = (const float4*)(hp + (long)c * 128);
    float4 v = src[lane];                      // coalesced 512B gather per wave

    float* dst = out + (long)r * 384 + layer * 128 + lane * 4;
    atomicAdd(dst + 0, v.x * nrm);
    atomicAdd(dst + 1, v.y * nrm);
    atomicAdd(dst + 2, v.z * nrm);
    atomicAdd(dst + 3, v.w * nrm);
}

// ---------------- launcher ----------------

extern "C" void kernel_launch(void* const* d_in, const int* in_sizes, int n_in,
                              void* d_out, int out_size, void* d_ws, size_t ws_size,
                              hipStream_t stream) {
    const float* x    = (const float*)d_in[0];
    const int*   edge = (const int*)d_in[1];
    const float* W0   = (const float*)d_in[2];
    const float* b0   = (const float*)d_in[3];
    const float* W1   = (const float*)d_in[4];
    const float* b1   = (const float*)d_in[5];
    const float* W2   = (const float*)d_in[6];
    const float* b2   = (const float*)d_in[7];
    float* out = (float*)d_out;

    const int n = in_sizes[0] / D;   // 100000 nodes
    const int e = in_sizes[1] / 2;   // 1600000 edges
    const int* rows = edge;          // edge_index[0] = destination
    const int* cols = edge + e;      // edge_index[1] = source

    // workspace layout: [dis: n floats][deg: n ints][h': n*128 floats]
    char* ws = (char*)d_ws;
    size_t seg = (((size_t)n * 4) + 255) & ~(size_t)255;
    float* dis = (float*)ws;
    int*   deg = (int*)(ws + seg);
    float* hp  = (float*)(ws + 2 * seg);

    deg_init_kernel<<<(n + 255) / 256, 256, 0, stream>>>(deg, n);
    deg_acc_kernel<<<(e + 255) / 256, 256, 0, stream>>>(cols, deg, e);
    dis_kernel<<<(n + 255) / 256, 256, 0, stream>>>(deg, dis, n);

    const float* Ws[3] = {W0, W1, W2};
    const float* bs[3] = {b0, b1, b2};

    for (int l = 0; l < 3; l++) {
        const float* hin  = (l == 0) ? x : (out + (l - 1) * 128);
        long in_stride    = (l == 0) ? 128 : 384;
        int  relu         = (l != 0);

        gemm_wmma_kernel<<<(n + 15) / 16, 256, 0, stream>>>(hin, in_stride, relu,
                                                            Ws[l], hp, n);
        selfloop_bias_kernel<<<(n + 1) / 2, 256, 0, stream>>>(hp, dis, bs[l], out, l, n);
        scatter_kernel<<<(e + 7) / 8, 256, 0, stream>>>(rows, cols, hp, dis, out, l, e);
    }
}